// SAGENet_73641509257822
// MI455X (gfx1250) — compile-verified
//
#include <hip/hip_runtime.h>

typedef __attribute__((ext_vector_type(2))) float v2f;
typedef __attribute__((ext_vector_type(8))) float v8f;

#define N_NODES 50000
#define N_EDGES 800000
#define IN_F    64
#define HID_F   64
#define OUT_F   32

// ---------------------------------------------------------------------------
// Zero-init a float region (grid-stride).
// ---------------------------------------------------------------------------
__global__ void zero_kernel(float* __restrict__ p, int n) {
    int i = blockIdx.x * blockDim.x + threadIdx.x;
    int stride = gridDim.x * blockDim.x;
    for (; i < n; i += stride) p[i] = 0.0f;
}

// ---------------------------------------------------------------------------
// Scatter-sum messages: 16 threads per edge, 4 features (one float4) each.
// Gather: one global_load_b128 per thread (coalesced 256B per edge row;
// all of X is L2-resident). Scatter: 4x global_atomic_add_f32 resolved at L2.
// DO_DEG folds the in-degree accumulation into the first scatter pass.
// ---------------------------------------------------------------------------
template <bool HAS_W, bool DO_DEG>
__global__ void scatter_kernel(const float* __restrict__ X,
                               const int* __restrict__ src,
                               const int* __restrict__ dst,
                               const float* __restrict__ w,
                               float* __restrict__ agg,
                               float* __restrict__ deg, int nE) {
    int t = blockIdx.x * blockDim.x + threadIdx.x;
    int e = t >> 4;          // edge index (16 threads per edge)
    int q = t & 15;          // feature quad: features [4q, 4q+3]
    if (e >= nE) return;
    int s = src[e];
    int d = dst[e];
    const float4 v4 = *(const float4*)(X + s * 64 + q * 4);
    float scale = HAS_W ? w[e] : 1.0f;
    float* ap = agg + d * 64 + q * 4;
    unsafeAtomicAdd(ap + 0, v4.x * scale);
    unsafeAtomicAdd(ap + 1, v4.y * scale);
    unsafeAtomicAdd(ap + 2, v4.z * scale);
    unsafeAtomicAdd(ap + 3, v4.w * scale);
    if (DO_DEG && q == 0) unsafeAtomicAdd(deg + d, 1.0f);
}

// ---------------------------------------------------------------------------
// Fused SAGE layer:  Out = act( Xs@Ws + (An/deg)@Wn + bias )
// One wave computes one 16x16 output tile with V_WMMA_F32_16X16X4_F32.
//
// fp32 WMMA fragment layouts (wave32):
//   A (16x4): lane L<16 -> row M=L, v0=K0,v1=K1 ; lane 16+L -> row L, v0=K2,v1=K3
//   B (4x16): lane L<16 -> col N=L, v0=K0,v1=K1 ; lane 16+L -> col L, v0=K2,v1=K3
//   C/D (16x16): 8 VGPRs; VGPR i = row (i + 8*half), lane%16 = column.
// ---------------------------------------------------------------------------
template <int K, int NO, bool RELU>
__global__ void sage_layer_kernel(const float* __restrict__ Xs,   // [rows, K]
                                  const float* __restrict__ An,   // [rows, K] unnormalized agg
                                  const float* __restrict__ deg,  // [rows]
                                  const float* __restrict__ Ws,   // [K, NO]
                                  const float* __restrict__ Wn,   // [K, NO]
                                  const float* __restrict__ bias, // [NO]
                                  float* __restrict__ Out,        // [rows, NO]
                                  int tilesM) {
    const int tilesN = NO / 16;
    int wave  = blockIdx.x * (blockDim.x >> 5) + (threadIdx.x >> 5);
    int tileM = wave / tilesN;
    int tileN = wave % tilesN;
    if (tileM >= tilesM) return;   // wave-uniform guard: EXEC stays all-ones

    int lane = threadIdx.x & 31;
    int half = lane >> 4;          // 0: K pair {0,1}, 1: K pair {2,3}
    int lm   = lane & 15;
    int arow = tileM * 16 + lm;    // A-matrix row owned by this lane
    int bcol = tileN * 16 + lm;    // B/C/D column owned by this lane

    v8f acc = {};

    // ---- self path: Xs @ Ws ----
    {
        const float* ap = Xs + arow * K + 2 * half;
        const float* bp = Ws + 2 * half * NO + bcol;
#pragma unroll
        for (int k = 0; k < K; k += 4) {
            v2f a; a.x = ap[k];          a.y = ap[k + 1];
            v2f b; b.x = bp[k * NO];     b.y = bp[k * NO + NO];
            acc = __builtin_amdgcn_wmma_f32_16x16x4_f32(
                false, a, false, b, (short)0, acc, false, false);
        }
    }

    // ---- neighbor path: (An / deg) @ Wn  (mean normalization folded in) ----
    {
        float rdeg = 1.0f / fmaxf(deg[arow], 1.0f);
        const float* ap = An + arow * K + 2 * half;
        const float* bp = Wn + 2 * half * NO + bcol;
#pragma unroll
        for (int k = 0; k < K; k += 4) {
            v2f a; a.x = ap[k] * rdeg;   a.y = ap[k + 1] * rdeg;
            v2f b; b.x = bp[k * NO];     b.y = bp[k * NO + NO];
            acc = __builtin_amdgcn_wmma_f32_16x16x4_f32(
                false, a, false, b, (short)0, acc, false, false);
        }
    }

    // ---- epilogue: bias (+ReLU), store ----
    float bv = bias[bcol];
#pragma unroll
    for (int i = 0; i < 8; ++i) {
        float v = acc[i] + bv;
        if (RELU) v = fmaxf(v, 0.0f);
        Out[(tileM * 16 + i + 8 * half) * NO + bcol] = v;
    }
}

// ---------------------------------------------------------------------------
// Host launch
// ---------------------------------------------------------------------------
extern "C" void kernel_launch(void* const* d_in, const int* in_sizes, int n_in,
                              void* d_out, int out_size, void* d_ws, size_t ws_size,
                              hipStream_t stream) {
    const float* x    = (const float*)d_in[0];   // [N, 64]
    const int*   esrc = (const int*)  d_in[1];   // [E]
    const int*   edst = (const int*)  d_in[2];   // [E]
    const float* ew   = (const float*)d_in[3];   // [E]
    const float* Ws1  = (const float*)d_in[4];   // [64, 64]
    const float* Wn1  = (const float*)d_in[5];   // [64, 64]
    const float* b1   = (const float*)d_in[6];   // [64]
    const float* Ws2  = (const float*)d_in[7];   // [64, 32]
    const float* Wn2  = (const float*)d_in[8];   // [64, 32]
    const float* b2   = (const float*)d_in[9];   // [32]
    float*       out  = (float*)d_out;           // [N, 32]

    // Workspace layout: deg | agg1 | agg2 | h  (deg..agg2 zeroed each call)
    float* ws   = (float*)d_ws;
    float* deg  = ws;                              // N
    float* agg1 = deg  + N_NODES;                  // N*64
    float* agg2 = agg1 + N_NODES * HID_F;          // N*64
    float* h    = agg2 + N_NODES * HID_F;          // N*64 (fully overwritten)

    const int zeroN = N_NODES + 2 * N_NODES * HID_F;   // deg + agg1 + agg2
    zero_kernel<<<2048, 256, 0, stream>>>(ws, zeroN);

    // Layer 1 aggregation (edge-weighted) + in-degree, fused
    {
        long long threads = (long long)N_EDGES * 16;
        int blocks = (int)((threads + 255) / 256);   // 50000 blocks
        scatter_kernel<true, true><<<blocks, 256, 0, stream>>>(
            x, esrc, edst, ew, agg1, deg, N_EDGES);
    }

    // Layer 1 GEMM + ReLU : h = relu(x@Ws1 + (agg1/deg)@Wn1 + b1)
    {
        const int tilesM = N_NODES / 16;            // 3125 (exact)
        const int waves  = tilesM * (HID_F / 16);   // 12500
        int blocks = (waves + 7) / 8;               // 8 waves / 256-thread block
        sage_layer_kernel<IN_F, HID_F, true><<<blocks, 256, 0, stream>>>(
            x, agg1, deg, Ws1, Wn1, b1, h, tilesM);
    }

    // Layer 2 aggregation (unweighted)
    {
        long long threads = (long long)N_EDGES * 16;
        int blocks = (int)((threads + 255) / 256);
        scatter_kernel<false, false><<<blocks, 256, 0, stream>>>(
            h, esrc, edst, nullptr, agg2, deg, N_EDGES);
    }

    // Layer 2 GEMM : out = h@Ws2 + (agg2/deg)@Wn2 + b2
    {
        const int tilesM = N_NODES / 16;            // 3125
        const int waves  = tilesM * (OUT_F / 16);   // 6250
        int blocks = (waves + 7) / 8;
        sage_layer_kernel<HID_F, OUT_F, false><<<blocks, 256, 0, stream>>>(
            h, agg2, deg, Ws2, Wn2, b2, out, tilesM);
    }
}